// GCN_21165598834696
// MI455X (gfx1250) — compile-verified
//
#include <hip/hip_runtime.h>

// ---------------------------------------------------------------------------
// CDNA5 / gfx1250 GCN pipeline, fp32 end-to-end.
//  - Dense GEMMs: V_WMMA_F32_16X16X4_F32, A-tile staged in LDS (coalesced
//    global load -> ds_store, fragments via ds_load, bank-conflict-free pad).
//  - Edge aggregation: coalesced gather + global fp32 atomics (exact
//    reference semantics of jax segment_sum).
//  - Edge head: warp-per-edge gather with global_prefetch of next edge rows.
// Memory-bound workload (~2.5 GB irregular traffic vs ~6 GFLOP dense math).
// ---------------------------------------------------------------------------

typedef float v2f __attribute__((ext_vector_type(2)));
typedef float v8f __attribute__((ext_vector_type(8)));

__global__ void k_zero(float* __restrict__ p, int n) {
    int i = blockIdx.x * blockDim.x + threadIdx.x;
    if (i < n) p[i] = 0.0f;
}

__global__ void k_count(const int* __restrict__ dst, float* __restrict__ deg, int E) {
    int e = blockIdx.x * blockDim.x + threadIdx.x;
    if (e < E) atomicAdd(&deg[dst[e]], 1.0f);
}

__global__ void k_dinv(const float* __restrict__ deg, float* __restrict__ dinv, int n) {
    int i = blockIdx.x * blockDim.x + threadIdx.x;
    if (i < n) dinv[i] = rsqrtf(deg[i] + 1.0f);   // deg counted at dst, +1 self-loop
}

// ---------------------------------------------------------------------------
// fp32 WMMA GEMM: C[M,N] = op(A)[M,K] @ B[K,N], op = identity or relu.
// Block = 256 threads (8 wave32). Block tile = 16 (M) x 128 (N):
//   * the 16xK A-tile is loaded once, coalesced, into LDS (ReLU fused here),
//   * each wave computes one 16x16 N-tile, A-fragments from LDS.
// WMMA layouts per CDNA5 ISA:
//   A frag (v2f): lane L -> row M = L&15, K = 2*(L>>4) + {0,1}
//   B frag (v2f): lane L -> col N = L&15, same K pair
//   C/D  (v8f) : VGPR r  -> row r + 8*(L>>4), col L&15
// LDS row stride = K+2 floats so lanes 0..15 (rows 0..15, same k) hit
// distinct banks (2*row mod 64).
// ---------------------------------------------------------------------------
template <bool RELU_A, int K>
__global__ void k_gemm_wmma(const float* __restrict__ A, const float* __restrict__ B,
                            float* __restrict__ C, int M, int N) {
    constexpr int LDK = K + 2;
    __shared__ float As[16 * LDK];

    const int mt   = blockIdx.x;     // M/16 tile
    const int nb   = blockIdx.y;     // N/128 group
    const int tid  = threadIdx.x;
    const int wave = tid >> 5;
    const int lane = tid & 31;

    // ---- cooperative, coalesced A-tile load (+ fused ReLU) ----------------
    const float* __restrict__ Ag = A + (size_t)mt * 16 * K;
#pragma unroll
    for (int idx = tid; idx < 16 * K; idx += 256) {
        int r  = idx / K;
        int kk = idx - r * K;
        float v = Ag[idx];
        if (RELU_A) v = v > 0.0f ? v : 0.0f;
        As[r * LDK + kk] = v;
    }
    __syncthreads();

    const int nt  = nb * 8 + wave;   // this wave's 16-col N tile
    const int col = lane & 15;       // A row / B,C col owned by this lane
    const int kb  = (lane >> 4) << 1;

    const float* __restrict__ Arow = As + col * LDK + kb;
    const float* __restrict__ Bc   = B + (size_t)nt * 16 + col;

    v8f acc = {};
#pragma unroll
    for (int k = 0; k < K; k += 4) {
        v2f av; av.x = Arow[k];                      av.y = Arow[k + 1];
        v2f bv; bv.x = Bc[(size_t)(k + kb) * N];     bv.y = Bc[(size_t)(k + kb + 1) * N];
        // 8 args: (neg_a, A, neg_b, B, c_mod, C, reuse_a, reuse_b)
        acc = __builtin_amdgcn_wmma_f32_16x16x4_f32(
            false, av, false, bv, (short)0, acc, false, false);
    }

    float* __restrict__ Cp =
        C + (size_t)(mt * 16 + ((lane >> 4) << 3)) * N + (size_t)nt * 16 + col;
#pragma unroll
    for (int r = 0; r < 8; ++r) Cp[(size_t)r * N] = acc[r];
}

// agg[i,f] = xw[i,f] * dinv[i]^2 + b[f]   (self-loop + bias folded into init)
__global__ void k_selfloop_bias(const float* __restrict__ xw, const float* __restrict__ dinv,
                                const float* __restrict__ b, float* __restrict__ agg,
                                int n, int F) {
    int idx = blockIdx.x * blockDim.x + threadIdx.x;
    if (idx < n * F) {
        int i = idx / F;
        int f = idx - i * F;
        float di = dinv[i];
        agg[idx] = xw[idx] * di * di + b[f];
    }
}

// One block per edge (blockDim == F): agg[dst] += dinv[src]*dinv[dst]*xw[src]
// src[e]/dst[e] are blockIdx-uniform -> scalarized to s_load by the compiler.
__global__ void k_scatter(const int* __restrict__ src, const int* __restrict__ dst,
                          const float* __restrict__ dinv, const float* __restrict__ xw,
                          float* __restrict__ agg, int F) {
    int e = blockIdx.x;
    int f = threadIdx.x;
    int s = src[e], d = dst[e];
    float c = dinv[s] * dinv[d];
    atomicAdd(&agg[(size_t)d * F + f], xw[(size_t)s * F + f] * c);
}

// node_x[i,:2] = emb[i,:128] @ Wn[128,2] + bn
__global__ void k_node_head(const float* __restrict__ emb, const float* __restrict__ Wn,
                            const float* __restrict__ bn, float* __restrict__ out, int n) {
    int i = blockIdx.x * blockDim.x + threadIdx.x;
    if (i >= n) return;
    const float4* row = (const float4*)(emb + (size_t)i * 128);
    float a0 = bn[0], a1 = bn[1];
#pragma unroll 8
    for (int q = 0; q < 32; ++q) {
        float4 v = row[q];
        int f = q * 4;
        a0 += v.x * Wn[(f + 0) * 2 + 0] + v.y * Wn[(f + 1) * 2 + 0] +
              v.z * Wn[(f + 2) * 2 + 0] + v.w * Wn[(f + 3) * 2 + 0];
        a1 += v.x * Wn[(f + 0) * 2 + 1] + v.y * Wn[(f + 1) * 2 + 1] +
              v.z * Wn[(f + 2) * 2 + 1] + v.w * Wn[(f + 3) * 2 + 1];
    }
    out[(size_t)i * 2 + 0] = a0;
    out[(size_t)i * 2 + 1] = a1;
}

// edge_x[e,:2] = [emb[src], emb[dst]] @ We[256,2] + be; warp (wave32) per edge.
// Prefetches the next edge's gathered rows (global_prefetch_b8).
__global__ void k_edge_head(const int* __restrict__ src, const int* __restrict__ dst,
                            const float* __restrict__ emb, const float* __restrict__ We,
                            const float* __restrict__ be, float* __restrict__ out, int E) {
    int lane = threadIdx.x & 31;
    int warp = blockIdx.x * (blockDim.x >> 5) + (threadIdx.x >> 5);
    int nwarps = gridDim.x * (blockDim.x >> 5);

    // lane owns features [4*lane, 4*lane+3] of the src half and dst half
    float ws0[4], ws1[4], wd0[4], wd1[4];
#pragma unroll
    for (int j = 0; j < 4; ++j) {
        int f = lane * 4 + j;
        ws0[j] = We[f * 2 + 0];         ws1[j] = We[f * 2 + 1];
        wd0[j] = We[(128 + f) * 2 + 0]; wd1[j] = We[(128 + f) * 2 + 1];
    }
    float b0 = be[0], b1 = be[1];

    for (int e = warp; e < E; e += nwarps) {
        int s = src[e], d = dst[e];
        int en = e + nwarps;
        if (en < E) {   // pull next iteration's gather rows toward the caches
            __builtin_prefetch(&emb[(size_t)src[en] * 128 + lane * 4], 0, 0);
            __builtin_prefetch(&emb[(size_t)dst[en] * 128 + lane * 4], 0, 0);
        }
        float4 vs = *(const float4*)(emb + (size_t)s * 128 + lane * 4);
        float4 vd = *(const float4*)(emb + (size_t)d * 128 + lane * 4);
        float a0 = vs.x * ws0[0] + vs.y * ws0[1] + vs.z * ws0[2] + vs.w * ws0[3] +
                   vd.x * wd0[0] + vd.y * wd0[1] + vd.z * wd0[2] + vd.w * wd0[3];
        float a1 = vs.x * ws1[0] + vs.y * ws1[1] + vs.z * ws1[2] + vs.w * ws1[3] +
                   vd.x * wd1[0] + vd.y * wd1[1] + vd.z * wd1[2] + vd.w * wd1[3];
#pragma unroll
        for (int off = 16; off > 0; off >>= 1) {
            a0 += __shfl_down(a0, off, 32);
            a1 += __shfl_down(a1, off, 32);
        }
        if (lane == 0) {
            out[(size_t)e * 2 + 0] = a0 + b0;
            out[(size_t)e * 2 + 1] = a1 + b1;
        }
    }
}

extern "C" void kernel_launch(void* const* d_in, const int* in_sizes, int n_in,
                              void* d_out, int out_size, void* d_ws, size_t ws_size,
                              hipStream_t stream) {
    const float* x  = (const float*)d_in[0];
    const int*   ei = (const int*)d_in[1];
    const float* W1 = (const float*)d_in[2];
    const float* b1 = (const float*)d_in[3];
    const float* W2 = (const float*)d_in[4];
    const float* b2 = (const float*)d_in[5];
    const float* Wn = (const float*)d_in[6];
    const float* bn = (const float*)d_in[7];
    const float* We = (const float*)d_in[8];
    const float* be = (const float*)d_in[9];

    const int N = in_sizes[0] / 64;   // 50000 (multiple of 16)
    const int E = in_sizes[1] / 2;    // 800000
    const int* src = ei;
    const int* dst = ei + E;

    // ---- workspace carve (256B aligned slabs) -----------------------------
    char*  ws  = (char*)d_ws;
    size_t off = 0;
    auto carve = [&](size_t bytes) -> float* {
        float* p = (float*)(ws + off);
        off += (bytes + 255) & ~(size_t)255;
        return p;
    };
    float* deg  = carve((size_t)N * 4);
    float* dinv = carve((size_t)N * 4);
    float* xw1  = carve((size_t)N * 256 * 4);
    float* agg1 = carve((size_t)N * 256 * 4);
    float* xw2  = carve((size_t)N * 128 * 4);
    float* emb  = carve((size_t)N * 128 * 4);

    float* out_node = (float*)d_out;            // [N,2]
    float* out_edge = out_node + (size_t)N * 2; // [E,2]

    // ---- degrees / normalization ------------------------------------------
    k_zero <<<(N + 255) / 256, 256, 0, stream>>>(deg, N);
    k_count<<<(E + 255) / 256, 256, 0, stream>>>(dst, deg, E);
    k_dinv <<<(N + 255) / 256, 256, 0, stream>>>(deg, dinv, N);

    // ---- layer 1: xw1 = x @ W1 (WMMA fp32, K=64), then GCN aggregate ------
    k_gemm_wmma<false, 64><<<dim3(N / 16, 256 / 128), 256, 0, stream>>>(x, W1, xw1, N, 256);
    k_selfloop_bias<<<(N * 256 + 255) / 256, 256, 0, stream>>>(xw1, dinv, b1, agg1, N, 256);
    k_scatter<<<E, 256, 0, stream>>>(src, dst, dinv, xw1, agg1, 256);

    // ---- layer 2: xw2 = relu(agg1) @ W2 (ReLU fused into LDS staging) -----
    k_gemm_wmma<true, 256><<<dim3(N / 16, 128 / 128), 256, 0, stream>>>(agg1, W2, xw2, N, 128);
    k_selfloop_bias<<<(N * 128 + 255) / 256, 256, 0, stream>>>(xw2, dinv, b2, emb, N, 128);
    k_scatter<<<E, 128, 0, stream>>>(src, dst, dinv, xw2, emb, 128);

    // ---- heads ------------------------------------------------------------
    k_node_head<<<(N + 255) / 256, 256, 0, stream>>>(emb, Wn, bn, out_node, N);
    k_edge_head<<<(E + 7) / 8, 256, 0, stream>>>(src, dst, emb, We, be, out_edge, E);
}